// temporal_graph_generator_17489106829579
// MI455X (gfx1250) — compile-verified
//
#include <hip/hip_runtime.h>

typedef __attribute__((ext_vector_type(2))) float v2f;
typedef __attribute__((ext_vector_type(8))) float v8f;

#define A_  32
#define H_  2048
// P layout: P[a][e][b][c], flat index ((a*32+e)*32+b)*32+c  -> 32*32*32*32 f32 = 4 MB

// ---------------------------------------------------------------------------
// Kernel 1: P[a,e,b,c] = sum_g T2[c,g] * ( sum_f T1[b,f] * ( sum_d nt[a,d]*k[d,e,f,g] ) )
// One block per (a,e) pair. ~200 MFLOP total; VALU is plenty.
// ---------------------------------------------------------------------------
__global__ __launch_bounds__(256) void build_P_kernel(
    const int*   __restrict__ t,
    const float* __restrict__ nodetime,
    const float* __restrict__ timevec1,
    const float* __restrict__ timevec2,
    const float* __restrict__ k,
    float*       __restrict__ P)
{
    const int blk = blockIdx.x;      // a*32 + e
    const int a   = blk >> 5;
    const int e   = blk & 31;
    const int tid = threadIdx.x;

    __shared__ float nt[32];
    __shared__ float M[32][33];      // adp1[a,e,:,:]  (f,g)
    __shared__ float Y[32][33];      // T1 @ M         (b,g)

    if (tid < 32) nt[tid] = nodetime[t[a] * 32 + tid];
    __syncthreads();

    // M[f][g] = sum_d nt[d] * k[d,e,f,g]
    for (int idx = tid; idx < 1024; idx += 256) {
        const int f = idx >> 5, g = idx & 31;
        float acc = 0.f;
        #pragma unroll
        for (int d = 0; d < 32; ++d)
            acc = fmaf(nt[d], k[((d * 32 + e) * 32 + f) * 32 + g], acc);
        M[f][g] = acc;
    }
    __syncthreads();

    // Y[b][g] = sum_f T1[b,f] * M[f][g]
    for (int idx = tid; idx < 1024; idx += 256) {
        const int b = idx >> 5, g = idx & 31;
        float acc = 0.f;
        #pragma unroll
        for (int f = 0; f < 32; ++f)
            acc = fmaf(timevec1[b * 32 + f], M[f][g], acc);
        Y[b][g] = acc;
    }
    __syncthreads();

    // P[a,e,b,c] = sum_g Y[b][g] * T2[c,g]
    for (int idx = tid; idx < 1024; idx += 256) {
        const int b = idx >> 5, c = idx & 31;
        float acc = 0.f;
        #pragma unroll
        for (int g = 0; g < 32; ++g)
            acc = fmaf(Y[b][g], timevec2[c * 32 + g], acc);
        P[(blk * 32 + b) * 32 + c] = acc;
    }
}

// ---------------------------------------------------------------------------
// Kernel 2: out[a,h,b,c] = softmax_c( relu( sum_e nodenum[h,e] * P[a,e,b,c] ) )
// One wave per (a, 16-row h-tile, b): 16x32 output tile = two f32 WMMA accumulators.
// K = 32 (e), chained V_WMMA_F32_16X16X4_F32, K-step = 4.
// ---------------------------------------------------------------------------
__global__ __launch_bounds__(256) void gemm_softmax_kernel(
    const float* __restrict__ nodenum,
    const float* __restrict__ P,
    float*       __restrict__ out)
{
    const int lane = threadIdx.x & 31;
    const int wid  = blockIdx.x * 8 + (threadIdx.x >> 5);
    const int b    = wid & 31;            // which 32-wide c-slab (the "b" index)
    const int ht   = (wid >> 5) & 127;    // h tile (16 rows each)
    const int a    = wid >> 12;           // batch index
    const int h0   = ht << 4;
    const int m    = lane & 15;           // row (A) / col (B,D) within tile
    const int khi  = lane >> 4;           // which half of the K=4 slice / row+8 for D

    // A: nodenum rows h0..h0+15, element A[m][k] in lane (m + 16*(k>>1)), vgpr (k&1)
    const float* __restrict__ Arow = nodenum + (h0 + m) * 32;
    // B: P[a][e][b][c], e is K, c is N; element B[k][n] in lane (n + 16*(k>>1)), vgpr (k&1)
    const float* __restrict__ Pb   = P + a * (32 * 1024) + b * 32;   // + e*1024 + c

    v8f acc0 = {0.f, 0.f, 0.f, 0.f, 0.f, 0.f, 0.f, 0.f};  // cols c = 0..15
    v8f acc1 = {0.f, 0.f, 0.f, 0.f, 0.f, 0.f, 0.f, 0.f};  // cols c = 16..31

    #pragma unroll
    for (int kk = 0; kk < 8; ++kk) {
        const int e0 = kk * 4 + khi * 2;             // this lane's K pair
        const v2f av = *(const v2f*)(Arow + e0);     // A[m][e0], A[m][e0+1]
        v2f b0, b1;
        b0.x = Pb[e0 * 1024 + m];
        b0.y = Pb[(e0 + 1) * 1024 + m];
        b1.x = Pb[e0 * 1024 + 16 + m];
        b1.y = Pb[(e0 + 1) * 1024 + 16 + m];
        acc0 = __builtin_amdgcn_wmma_f32_16x16x4_f32(false, av, false, b0,
                                                     (short)0, acc0, false, false);
        acc1 = __builtin_amdgcn_wmma_f32_16x16x4_f32(false, av, false, b1,
                                                     (short)0, acc1, false, false);
    }

    // D layout: VGPR r holds row (r + 8*khi), col = m. Softmax over c = 0..31:
    // 16-lane half-wave reductions via shfl_xor masks 1,2,4,8 (stay within half).
    float* __restrict__ obase =
        out + ((long)a * H_ + h0 + 8 * khi) * 1024 + b * 32;

    #pragma unroll
    for (int r = 0; r < 8; ++r) {
        float x0 = fmaxf(acc0[r], 0.f);
        float x1 = fmaxf(acc1[r], 0.f);
        float mx = fmaxf(x0, x1);
        mx = fmaxf(mx, __shfl_xor(mx, 1, 32));
        mx = fmaxf(mx, __shfl_xor(mx, 2, 32));
        mx = fmaxf(mx, __shfl_xor(mx, 4, 32));
        mx = fmaxf(mx, __shfl_xor(mx, 8, 32));
        float ex0 = __expf(x0 - mx);
        float ex1 = __expf(x1 - mx);
        float s = ex0 + ex1;
        s += __shfl_xor(s, 1, 32);
        s += __shfl_xor(s, 2, 32);
        s += __shfl_xor(s, 4, 32);
        s += __shfl_xor(s, 8, 32);
        const float inv = 1.0f / s;
        float* orow = obase + r * 1024;
        __builtin_nontemporal_store(ex0 * inv, orow + m);        // c = m
        __builtin_nontemporal_store(ex1 * inv, orow + m + 16);   // c = m + 16
    }
}

// ---------------------------------------------------------------------------
extern "C" void kernel_launch(void* const* d_in, const int* in_sizes, int n_in,
                              void* d_out, int out_size, void* d_ws, size_t ws_size,
                              hipStream_t stream) {
    (void)in_sizes; (void)n_in; (void)out_size; (void)ws_size;
    const int*   t        = (const int*)  d_in[0];
    const float* nodetime = (const float*)d_in[1];
    const float* nodenum  = (const float*)d_in[2];
    const float* timevec1 = (const float*)d_in[3];
    const float* timevec2 = (const float*)d_in[4];
    const float* k        = (const float*)d_in[5];
    float* P = (float*)d_ws;   // 4 MB scratch: P[32][32][32][32]

    build_P_kernel<<<1024, 256, 0, stream>>>(t, nodetime, timevec1, timevec2, k, P);
    // 32 a * 128 h-tiles * 32 b = 131072 waves / 8 waves per block
    gemm_softmax_kernel<<<16384, 256, 0, stream>>>(nodenum, P, (float*)d_out);
}